// BasicGenerativeDeconvolutionBlock_40699110096960
// MI455X (gfx1250) — compile-verified
//
#include <hip/hip_runtime.h>
#include <hip/hip_bf16.h>

typedef __attribute__((ext_vector_type(2))) float v2f;
typedef __attribute__((ext_vector_type(8))) float v8f;

#define INC 64
#define OUTC 64
#define KTAPS 27
#define EPS 1e-5f
#define NEG_SLOPE 0.01f

// ---------------------------------------------------------------------------
// Kernel 0: zero output accumulator + per-channel stats buffer (128 floats)
// ---------------------------------------------------------------------------
__global__ void init_kernel(float* __restrict__ out, long total, float* __restrict__ stats) {
    long i = (long)blockIdx.x * blockDim.x + threadIdx.x;
    long stride = (long)gridDim.x * blockDim.x;
    for (; i < total; i += stride) out[i] = 0.0f;
    long t = (long)blockIdx.x * blockDim.x + threadIdx.x;
    if (t < 128) stats[t] = 0.0f;
}

// ---------------------------------------------------------------------------
// Wave body: one wave32 handles 16 input rows (rowbase..rowbase+15).
// A-frag layout (ISA 7.12.2, 32-bit A 16x4):
//   lane L (L<16): row = L, VGPR0=K0, VGPR1=K1 ; lane L+16: same row, K2/K3.
// B-frag (4x16): lanes hold N=lane&15; half 0 -> K rows {0,1}, half 1 -> {2,3}.
// C layout: VGPR j -> row j + 8*half, col = lane&15.
// FULL=true: all 16 rows valid -> no per-element guards anywhere.
// All scatter offsets fit in u32 (out buffer < 4 GB), so atomics can use the
// saddr + 32-bit voffset addressing form.
// ---------------------------------------------------------------------------
template <bool FULL>
__device__ __forceinline__ void
wave_tile(const float* __restrict__ feats,
          const float* __restrict__ W,
          const int*   __restrict__ out_idx,
          float*       __restrict__ out,
          int rowbase, int N, int lane) {
    const int half = lane >> 4;     // 0: K pair {0,1}, 1: K pair {2,3}
    const int l16  = lane & 15;

    const int  row = rowbase + l16;
    const bool rv  = FULL || (row < N);

    // Load the 16x64 A tile once: a[kk] covers K = 4*kk + 2*half + {0,1}
    v2f a[16];
    const float* fr = feats + (long)row * INC + 2 * half;
#pragma unroll
    for (int kk = 0; kk < 16; ++kk) {
        if (FULL) {
            a[kk] = *(const v2f*)(fr + kk * 4);
        } else {
            v2f av; av.x = 0.0f; av.y = 0.0f;
            if (rv) av = *(const v2f*)(fr + kk * 4);
            a[kk] = av;
        }
    }

    const unsigned idx_base = (unsigned)row * KTAPS;  // u32 gather offsets

    for (int k = 0; k < KTAPS; ++k) {
        // destination row for this lane's input row (lanes 0..15 == 16..31)
        int ridx = -1;
        if (rv) ridx = out_idx[idx_base + k];

        // Hoist the 8 row-index shuffles out of the N-tile loop: lane needs
        // destinations of C rows (j + 8*half) for j = 0..7.
        unsigned obase[8];
#pragma unroll
        for (int j = 0; j < 8; ++j)
            obase[j] = (unsigned)__shfl(ridx, j + 8 * half, 32) * (unsigned)OUTC;

        const float* wk = W + (long)k * INC * OUTC;
#pragma unroll
        for (int nt = 0; nt < 4; ++nt) {
            v8f c = {};
            const float* wb = wk + nt * 16 + l16;
#pragma unroll
            for (int kk = 0; kk < 16; ++kk) {
                const int krow = kk * 4 + 2 * half;
                v2f b;
                b.x = wb[(long)krow * OUTC];
                b.y = wb[(long)(krow + 1) * OUTC];
                c = __builtin_amdgcn_wmma_f32_16x16x4_f32(
                        /*neg_a=*/false, a[kk],
                        /*neg_b=*/false, b,
                        /*c_mod=*/(short)0, c,
                        /*reuse_a=*/false, /*reuse_b=*/false);
            }
            // Scatter 16x16 C tile: lane holds rows (j + 8*half), col nt*16+l16
            const unsigned col = (unsigned)(nt * 16 + l16);
#pragma unroll
            for (int j = 0; j < 8; ++j) {
                if (FULL) {
                    atomicAdd(out + (obase[j] + col), c[j]);
                } else if ((int)obase[j] >= 0) {   // invalid rows: ridx = -1
                    atomicAdd(out + (obase[j] + col), c[j]);
                }
            }
        }
    }
}

// ---------------------------------------------------------------------------
// Kernel 1: per-16-row-tile WMMA GEMM (f32 16x16x4) + atomic scatter.
// 8 waves/block, 16 rows per wave; nearly all tiles take the FULL fast path.
// ---------------------------------------------------------------------------
__global__ void __launch_bounds__(256)
scatter_gemm_kernel(const float* __restrict__ feats,
                    const float* __restrict__ W,
                    const int*   __restrict__ out_idx,
                    float*       __restrict__ out,
                    int N) {
    const int wave = threadIdx.x >> 5;
    const int lane = threadIdx.x & 31;
    const int rowbase = (blockIdx.x * 8 + wave) * 16;
    if (rowbase >= N) return;       // wave-uniform exit

    if (rowbase + 16 <= N) {
        wave_tile<true>(feats, W, out_idx, out, rowbase, N, lane);
    } else {
        wave_tile<false>(feats, W, out_idx, out, rowbase, N, lane);
    }
}

// ---------------------------------------------------------------------------
// Kernel 2: per-channel sum / sumsq reduction. blockDim=256, stride is a
// multiple of 64 so each thread owns a fixed channel (tid & 63).
// ---------------------------------------------------------------------------
__global__ void __launch_bounds__(256)
bn_stats_kernel(const float* __restrict__ out, float* __restrict__ stats, long total) {
    float s = 0.0f, s2 = 0.0f;
    long i = (long)blockIdx.x * blockDim.x + threadIdx.x;
    const long stride = (long)gridDim.x * blockDim.x;
    for (; i < total; i += stride) {
        const float v = out[i];
        s += v; s2 += v * v;
    }
    __shared__ float ls[256], lq[256];
    ls[threadIdx.x] = s; lq[threadIdx.x] = s2;
    __syncthreads();
    if (threadIdx.x < 64) {
        const int c = threadIdx.x;
        s  = ls[c] + ls[c + 64] + ls[c + 128] + ls[c + 192];
        s2 = lq[c] + lq[c + 64] + lq[c + 128] + lq[c + 192];
        atomicAdd(&stats[c], s);
        atomicAdd(&stats[64 + c], s2);
    }
}

// ---------------------------------------------------------------------------
// Kernel 3: y = gamma * (x - mean) * rsqrt(var + eps) + beta, then LeakyReLU.
// ---------------------------------------------------------------------------
__global__ void __launch_bounds__(256)
bn_apply_kernel(float* __restrict__ out, const float* __restrict__ stats,
                const float* __restrict__ gamma, const float* __restrict__ beta,
                long total, float invN) {
    long i = (long)blockIdx.x * blockDim.x + threadIdx.x;
    const long stride = (long)gridDim.x * blockDim.x;
    for (; i < total; i += stride) {
        const int c = (int)(i & 63);
        const float mean = stats[c] * invN;
        const float var  = stats[64 + c] * invN - mean * mean;
        const float sc   = gamma[c] * rsqrtf(var + EPS);
        const float y    = (out[i] - mean) * sc + beta[c];
        out[i] = y > 0.0f ? y : NEG_SLOPE * y;
    }
}

extern "C" void kernel_launch(void* const* d_in, const int* in_sizes, int n_in,
                              void* d_out, int out_size, void* d_ws, size_t ws_size,
                              hipStream_t stream) {
    // setup_inputs order: coords, feats, W, gamma, beta, out_idx, out_template
    const float* feats   = (const float*)d_in[1];
    const float* W       = (const float*)d_in[2];
    const float* gamma   = (const float*)d_in[3];
    const float* beta    = (const float*)d_in[4];
    const int*   out_idx = (const int*)d_in[5];

    float* out   = (float*)d_out;
    float* stats = (float*)d_ws;          // 128 floats: [sum(64) | sumsq(64)]

    const int  N     = in_sizes[1] / INC; // 100000
    const long total = (long)out_size;    // N_out * 64
    const long n_out = total / OUTC;

    // 0) zero accumulator + stats
    {
        long blocks = (total + 255) / 256;
        if (blocks > 262144) blocks = 262144;
        init_kernel<<<dim3((unsigned)blocks), dim3(256), 0, stream>>>(out, total, stats);
    }
    // 1) WMMA GEMM + atomic scatter: 8 waves/block, 16 rows per wave
    {
        const int blocks = (N + 127) / 128;
        scatter_gemm_kernel<<<dim3(blocks), dim3(256), 0, stream>>>(feats, W, out_idx, out, N);
    }
    // 2) per-channel stats
    bn_stats_kernel<<<dim3(2048), dim3(256), 0, stream>>>(out, stats, total);
    // 3) normalize + LeakyReLU
    {
        long blocks = (total + 255) / 256;
        if (blocks > 262144) blocks = 262144;
        bn_apply_kernel<<<dim3((unsigned)blocks), dim3(256), 0, stream>>>(
            out, stats, gamma, beta, total, 1.0f / (float)n_out);
    }
}